// BinomialResample_59571196396024
// MI455X (gfx1250) — compile-verified
//
#include <hip/hip_runtime.h>
#include <stdint.h>

// 4-wide float vector (clang ext vector so __builtin_nontemporal_store works)
typedef __attribute__((ext_vector_type(4))) float v4f;

#define THREADS 256
#define GENES 20000u   // columns of x_ng (reference shape 8192 x 20000)

__device__ __forceinline__ uint32_t pcg_hash(uint32_t v) {
    v = v * 747796405u + 2891336453u;
    uint32_t w = ((v >> ((v >> 28u) + 4u)) ^ v) * 277803737u;
    return (w >> 22u) ^ w;
}

// Per-float4 RNG pre-work (independent of x): apply mask, 16-bit p-thresholds, trial seeds.
struct Pre {
    uint32_t thr[4];
    uint32_t st[4];
    bool     apply;
};

__device__ __forceinline__ Pre make_pre(uint32_t idx4) {
    Pre pr;
    const uint32_t base = idx4 * 4u;
    const uint32_t row  = base / GENES;            // const divisor -> mulhi; all 4 elems share a row
    pr.apply = (pcg_hash(row ^ 0x9E3779B9u) & 1u) != 0u;
#pragma unroll
    for (int j = 0; j < 4; ++j) {
        const uint32_t e  = base + (uint32_t)j;
        const uint32_t hp = pcg_hash(e ^ 0x85EBCA6Bu);
        pr.thr[j] = 32768u + (uint32_t)(((uint64_t)hp * 26214u) >> 32);   // p*65536, p in [0.5,0.9)
        pr.st[j]  = pcg_hash(e ^ 0xC2B2AE35u);
    }
    return pr;
}

// Binomial(n=x[j], p) via paired 16-bit Bernoulli trials; passthrough when !apply.
__device__ __forceinline__ v4f binom_chunk(const v4f xv, const Pre& pr) {
    v4f out;
#pragma unroll
    for (int j = 0; j < 4; ++j) {
        const float xf = xv[j];
        if (pr.apply) {
            const int n = (int)xf;                 // integer-valued UMI count, 0..49
            uint32_t s = pr.st[j], sum = 0;
            const uint32_t t16 = pr.thr[j];
            const int pairs = n >> 1;
            for (int k = 0; k < pairs; ++k) {      // 2 Bernoulli(p) trials per 32-bit draw
                s = s * 1664525u + 1013904223u;
                const uint32_t w = s ^ (s >> 16);
                sum += ((w & 0xFFFFu) < t16);
                sum += ((w >> 16)     < t16);
            }
            if (n & 1) {
                s = s * 1664525u + 1013904223u;
                sum += (((s ^ (s >> 16)) & 0xFFFFu) < t16);
            }
            out[j] = (float)sum;
        } else {
            out[j] = xf;
        }
    }
    return out;
}

__global__ void __launch_bounds__(THREADS)
binom_resample_kernel(const float* __restrict__ x, float* __restrict__ y, int total4) {
    __shared__ v4f tile[2 * THREADS];

    const int tid = threadIdx.x;
    const int i0  = blockIdx.x * (2 * THREADS) + tid;       // chunk A
    if (i0 >= total4) return;                               // (wave-coherent: exact-fit grid)
    const int i1r = i0 + THREADS;                           // chunk B
    const bool ok1 = (i1r < total4);
    const int i1  = ok1 ? i1r : i0;                         // clamp: keep ASYNCcnt bookkeeping exact

    // ---- CDNA5 async DMA double-buffer: issue both 16B/lane copies up front (NT: streamed once)
    const uint32_t l0 = (uint32_t)(uintptr_t)(&tile[tid]);            // low 32b of generic == LDS offset
    const uint32_t l1 = (uint32_t)(uintptr_t)(&tile[tid + THREADS]);
    const uint64_t g0 = (uint64_t)(uintptr_t)(x + (size_t)i0 * 4);
    const uint64_t g1 = (uint64_t)(uintptr_t)(x + (size_t)i1 * 4);
    asm volatile("global_load_async_to_lds_b128 %0, %1, off th:TH_LOAD_NT"
                 :: "v"(l0), "v"(g0) : "memory");
    asm volatile("global_load_async_to_lds_b128 %0, %1, off th:TH_LOAD_NT"
                 :: "v"(l1), "v"(g1) : "memory");

    // ---- overlap: RNG pre-work for both chunks while the DMAs are in flight
    const Pre prA = make_pre((uint32_t)i0);
    const Pre prB = make_pre((uint32_t)i1r);

    // ---- partial wait: chunk A landed (async loads complete in order) ----
    asm volatile("s_wait_asynccnt 1" ::: "memory");
    {
        const v4f xv  = tile[tid];
        const v4f out = binom_chunk(xv, prA);
        __builtin_nontemporal_store(out, (v4f*)y + i0);     // write-once output, 3.4x L2 size
    }
    // (chunk B's DMA overlapped all of chunk A's compute + store above)

    // ---- drain: chunk B landed ----
    asm volatile("s_wait_asynccnt 0" ::: "memory");
    if (ok1) {
        const v4f xv  = tile[tid + THREADS];
        const v4f out = binom_chunk(xv, prB);
        __builtin_nontemporal_store(out, (v4f*)y + i1r);
    }
}

extern "C" void kernel_launch(void* const* d_in, const int* in_sizes, int n_in,
                              void* d_out, int out_size, void* d_ws, size_t ws_size,
                              hipStream_t stream) {
    const float* x = (const float*)d_in[0];
    float*       y = (float*)d_out;
    const int total  = in_sizes[0];                 // 8192 * 20000
    const int total4 = total / 4;                   // float4 granules
    const int per_block = 2 * THREADS;              // 2 chunks per thread (double buffer)
    const int blocks = (total4 + per_block - 1) / per_block;
    hipLaunchKernelGGL(binom_resample_kernel, dim3(blocks), dim3(THREADS), 0, stream,
                       x, y, total4);
}